// SpecAugment_61641370632593
// MI455X (gfx1250) — compile-verified
//
#include <hip/hip_runtime.h>
#include <cstdint>

// ---------------------------------------------------------------------------
// SpecAugment for MI455X (gfx1250).
//
// Roofline: 64*80*4000 f32 = 81.9 MB in + 81.9 MB out = 164 MB streaming at
// 23.3 TB/s => ~7.0 us floor. Arithmetic intensity ~0, so this is a pure
// data-movement kernel: b128 vector traffic, async global->LDS staging of the
// per-batch time mask (CDNA5 ASYNCcnt path), default RT cache hints since the
// whole working set fits in the 192 MB L2.
//
// RNG reproduces JAX threefry2x32 with jax_threefry_partitionable=True (the
// modern default): subkey_i = threefry(key, (0, i)); 32-bit draws are
// xor(hi, lo) of the threefry output pair; randint uses the two-draw
// multiply-mod scheme. Root key = jax.random.key(42) = (0, 42).
// ---------------------------------------------------------------------------

constexpr int kBatch     = 64;
constexpr int kFreq      = 80;
constexpr int kTime      = 4000;
constexpr int kFreqMasks = 2;
constexpr int kTimeMasks = 10;
constexpr int kFreqWidth = 27;
constexpr int kTimeWidth = 100;
constexpr int kT4        = kTime / 4;        // 1000 float4 per row
constexpr int kRows      = kBatch * kFreq;   // 5120 rows
constexpr int kTmaskOff  = 8192;             // ws byte offset of time mask

typedef int v4i __attribute__((vector_size(4 * sizeof(int))));
typedef v4i __attribute__((address_space(1)))* gv4i_p;   // global int4*
typedef v4i __attribute__((address_space(3)))* lv4i_p;   // LDS int4*

struct Key2 { uint32_t a, b; };

__device__ __forceinline__ void tf2x32(uint32_t k0, uint32_t k1,
                                       uint32_t c0, uint32_t c1,
                                       uint32_t& o0, uint32_t& o1) {
  const uint32_t ks2 = k0 ^ k1 ^ 0x1BD11BDAu;
  uint32_t x0 = c0 + k0;
  uint32_t x1 = c1 + k1;
#define TFR(r) { x0 += x1; x1 = (x1 << (r)) | (x1 >> (32 - (r))); x1 ^= x0; }
  TFR(13) TFR(15) TFR(26) TFR(6)   x0 += k1;  x1 += ks2 + 1u;
  TFR(17) TFR(29) TFR(16) TFR(24)  x0 += ks2; x1 += k0  + 2u;
  TFR(13) TFR(15) TFR(26) TFR(6)   x0 += k0;  x1 += k1  + 3u;
  TFR(17) TFR(29) TFR(16) TFR(24)  x0 += k1;  x1 += ks2 + 4u;
  TFR(13) TFR(15) TFR(26) TFR(6)   x0 += ks2; x1 += k0  + 5u;
#undef TFR
  o0 = x0; o1 = x1;
}

// jax.random.split(key)[i]  (partitionable / fold-like)
__device__ __forceinline__ Key2 foldsplit(Key2 k, uint32_t i) {
  Key2 r; tf2x32(k.a, k.b, 0u, i, r.a, r.b); return r;
}
// 32-bit random bits, flat element i (partitionable)
__device__ __forceinline__ uint32_t rbits(Key2 k, uint32_t i) {
  uint32_t a, b; tf2x32(k.a, k.b, 0u, i, a, b); return a ^ b;
}
// jax.random.randint(key, shape, 0, span) at flat index `flat`
__device__ __forceinline__ uint32_t jrandint(Key2 k, uint32_t flat, uint32_t span) {
  Key2 k1 = foldsplit(k, 0u);
  Key2 k2 = foldsplit(k, 1u);
  uint32_t hi = rbits(k1, flat);
  uint32_t lo = rbits(k2, flat);
  uint32_t m  = 1073741824u % span;   // 2^30 % span
  m = (m * m) % span;                 // (2^30 % span)^2 % span
  return ((hi % span) * m + (lo % span)) % span;
}

// One block per batch element: compute the 2 freq bands + 10 time bands, then
// expand to byte masks (1 = masked).
__global__ void specaug_maskgen(uint8_t* __restrict__ fmask,
                                uint8_t* __restrict__ tmask) {
  const int b   = blockIdx.x;
  const int tid = threadIdx.x;

  __shared__ int s_fs[kFreqMasks], s_fe[kFreqMasks];
  __shared__ int s_ts[kTimeMasks], s_te[kTimeMasks];

  if (tid < kFreqMasks) {
    const Key2 mkey = {0u, 42u};                // jax.random.key(42)
    const Key2 kf   = foldsplit(mkey, 0u);      // kf, kt = split(mkey)
    const Key2 ks   = foldsplit(kf, 0u);        // k_start, k_width = split(kf)
    const Key2 kw   = foldsplit(kf, 1u);
    const uint32_t flat = (uint32_t)(b * kFreqMasks + tid);
    const int st = (int)jrandint(ks, flat, (uint32_t)(kFreq - kFreqWidth)); // span 53
    const int wd = (int)jrandint(kw, flat, (uint32_t)kFreqWidth);           // span 27
    s_fs[tid] = st; s_fe[tid] = st + wd;
  } else if (tid < kFreqMasks + kTimeMasks) {
    const int m = tid - kFreqMasks;
    const Key2 mkey = {0u, 42u};
    const Key2 kt   = foldsplit(mkey, 1u);
    const Key2 ks   = foldsplit(kt, 0u);
    const Key2 kw   = foldsplit(kt, 1u);
    const uint32_t flat = (uint32_t)(b * kTimeMasks + m);
    const int st = (int)jrandint(ks, flat, (uint32_t)(kTime - kTimeWidth)); // span 3900
    const int wd = (int)jrandint(kw, flat, (uint32_t)kTimeWidth);           // span 100
    s_ts[m] = st; s_te[m] = st + wd;
  }
  __syncthreads();

  if (tid < kFreq) {
    bool msk = false;
#pragma unroll
    for (int j = 0; j < kFreqMasks; ++j)
      msk |= (tid >= s_fs[j]) && (tid < s_fe[j]);
    fmask[b * kFreq + tid] = msk ? 1u : 0u;
  }
  for (int t = tid; t < kTime; t += blockDim.x) {
    bool msk = false;
#pragma unroll
    for (int j = 0; j < kTimeMasks; ++j)
      msk |= (t >= s_ts[j]) && (t < s_te[j]);
    tmask[b * kTime + t] = msk ? 1u : 0u;
  }
}

// One block per (batch, freq) row of 4000 floats. Stage the batch's 4000-byte
// time mask into LDS via CDNA5 async global->LDS, then stream float4.
__global__ void specaug_apply(const float* __restrict__ in,
                              float* __restrict__ out,
                              const uint8_t* __restrict__ fmask,
                              uint8_t* __restrict__ tmask) {
  const int row = blockIdx.x;          // 0 .. kRows-1
  const int b   = row / kFreq;

  __shared__ __align__(16) uint32_t s_tm[kT4];  // 4000 bytes

  uint32_t* trow32 = (uint32_t*)(tmask + (size_t)b * kTime);  // 16B-aligned

  {
    const int i = threadIdx.x;
    if (i < kT4 / 4) {  // 250 b128 transfers
#if __has_builtin(__builtin_amdgcn_global_load_async_to_lds_b128)
      __builtin_amdgcn_global_load_async_to_lds_b128(
          (gv4i_p)(trow32 + i * 4),
          (lv4i_p)(&s_tm[i * 4]),
          /*offset=*/0, /*cpol=*/0);
#else
      asm volatile("global_load_async_to_lds_b128 %0, %1, off"
                   :
                   : "v"((uint32_t)(uintptr_t)(&s_tm[i * 4])),
                     "v"(trow32 + i * 4)
                   : "memory");
#endif
    }
#if __has_builtin(__builtin_amdgcn_s_wait_asynccnt)
    __builtin_amdgcn_s_wait_asynccnt(0);
#else
    asm volatile("s_wait_asynccnt 0" ::: "memory");
#endif
    __syncthreads();
  }

  const uint32_t fmw = fmask[row] ? 0xFFFFFFFFu : 0u;

  const float4* __restrict__ inrow  = (const float4*)(in  + (size_t)row * kTime);
  float4*       __restrict__ outrow = (float4*)(out + (size_t)row * kTime);

  for (int i = threadIdx.x; i < kT4; i += blockDim.x) {
    if (i + (int)blockDim.x < kT4)
      __builtin_prefetch(&inrow[i + blockDim.x], 0, 0);  // global_prefetch_b8
    float4 v = inrow[i];
    const uint32_t tm = s_tm[i] | fmw;
    v.x = (tm & 0x000000FFu) ? 0.0f : v.x;
    v.y = (tm & 0x0000FF00u) ? 0.0f : v.y;
    v.z = (tm & 0x00FF0000u) ? 0.0f : v.z;
    v.w = (tm & 0xFF000000u) ? 0.0f : v.w;
    outrow[i] = v;
  }
}

extern "C" void kernel_launch(void* const* d_in, const int* in_sizes, int n_in,
                              void* d_out, int out_size, void* d_ws, size_t ws_size,
                              hipStream_t stream) {
  (void)in_sizes; (void)n_in; (void)out_size; (void)ws_size;
  const float* in  = (const float*)d_in[0];
  float*       out = (float*)d_out;
  uint8_t* fmask = (uint8_t*)d_ws;                    // 64*80    = 5120 B
  uint8_t* tmask = (uint8_t*)d_ws + kTmaskOff;        // 64*4000  = 256000 B

  specaug_maskgen<<<kBatch, 128, 0, stream>>>(fmask, tmask);
  specaug_apply<<<kRows, 256, 0, stream>>>(in, out, fmask, tmask);
}